// WaveletLoss_55559696941629
// MI455X (gfx1250) — compile-verified
//
#include <hip/hip_runtime.h>

typedef __attribute__((ext_vector_type(2))) float v2f;
typedef __attribute__((ext_vector_type(8))) float v8f;

#define WTH   1024      // row length
#define M1    519       // level-1 output length  (1024+15)/2
#define M2    267       // level-2 output length  (519+15)/2
#define M3    141       // level-3 output length  (267+15)/2
#define ST1   1074      // LDS row stride (floats), >= 2+n+29 and covers tile overreads; == 2 (mod 4) -> conflict-free transposed reads
#define ST2   562
#define ST3   306
#define NTHR  512       // 16 wave32s

// pywt db8 dec_lo coefficients
__device__ __constant__ float cDB8LO[16] = {
    -0.00011747678400228192f,  0.0006754494059985568f, -0.0003917403729959771f,
    -0.00487035299301066f,     0.008746094047015655f,   0.013981027917015516f,
    -0.04408825393106472f,    -0.01736930100202211f,    0.128747426620186f,
     0.00047248457399797254f, -0.2840155429624281f,    -0.015829105256023893f,
     0.5853546836548691f,      0.6756307362980128f,     0.3128715909144659f,
     0.05441584224308161f
};

// Rebuild symmetric padding of a 16-row LDS buffer whose data occupies
// [16, 16+n) of each stride-length row. ext coordinate k maps to buffer[2+k].
// left pad (14): ext[k] = x[13-k]; right pad (15): ext[14+n+j] = x[n-1-j].
// Also zeroes [0,2) and the tail up to `stride` (tiles overread there).
__device__ __attribute__((always_inline))
void fill_pads(float* ex, int n, int stride, int tid)
{
    const int tail0 = 16 + n + 15;
    const int ztail = stride - tail0;
    const int per   = 14 + 15 + 2 + ztail;
    for (int idx = tid; idx < 16 * per; idx += NTHR) {
        const int row = idx / per;
        const int p   = idx % per;
        float* b = ex + row * stride;
        if (p < 14)          b[2 + p]          = b[16 + 13 - p];
        else if (p < 29)   { const int j = p - 14; b[16 + n + j] = b[16 + n - 1 - j]; }
        else if (p < 31)     b[p - 29]         = 0.0f;
        else                 b[tail0 + (p-31)] = 0.0f;
    }
}

// One DWT level for 16 rows: D[16 outputs x 16 rows] = sum_kb A(Toeplitz) x B(window).
// Accumulates weighted squared cD (hi) diffs into acc; optionally stores cA (lo)
// into the next level's ext buffer at data offset 16.
template <bool STORE>
__device__ __attribute__((always_inline))
void process_level(const float* __restrict__ exX, const float* __restrict__ exY,
                   float* __restrict__ outX, float* __restrict__ outY,
                   int m, int strideIn, int strideOut, float w, float& acc,
                   const v2f* Alo, const v2f* Ahi,
                   int mrow, int half, int wave, int nwaves)
{
    const int T = (m + 15) >> 4;
    // Transposed B reads: lane -> row (N dim), VGPR/half -> K dim.
    const float* bxBase = exX + mrow * strideIn + 2 * half + 2;
    const float* byBase = exY + mrow * strideIn + 2 * half + 2;
    float lsum = 0.0f;
    for (int tile = wave; tile < T; tile += nwaves) {   // wave-uniform: EXEC all-ones at WMMA
        const int base = 32 * tile;
        v8f dxlo = 0.0f, dxhi = 0.0f, dylo = 0.0f, dyhi = 0.0f;
#pragma unroll
        for (int kb = 0; kb < 12; ++kb) {               // K = 48 (>= 46), zero-padded in A
            v2f bx = *(const v2f*)(bxBase + base + kb * 4);
            v2f by = *(const v2f*)(byBase + base + kb * 4);
            dxlo = __builtin_amdgcn_wmma_f32_16x16x4_f32(false, Alo[kb], false, bx, (short)0, dxlo, false, false);
            dxhi = __builtin_amdgcn_wmma_f32_16x16x4_f32(false, Ahi[kb], false, bx, (short)0, dxhi, false, false);
            dylo = __builtin_amdgcn_wmma_f32_16x16x4_f32(false, Alo[kb], false, by, (short)0, dylo, false, false);
            dyhi = __builtin_amdgcn_wmma_f32_16x16x4_f32(false, Ahi[kb], false, by, (short)0, dyhi, false, false);
        }
        // D layout: lane holds N=lane&15 (row); VGPR v holds M = v + 8*(lane>>4) (output pos)
#pragma unroll
        for (int v = 0; v < 8; ++v) {
            const int i = 16 * tile + 8 * half + v;
            if (i < m) {
                const float d = dxhi[v] - dyhi[v];
                lsum += d * d;
                if (STORE) {
                    outX[mrow * strideOut + 16 + i] = dxlo[v];
                    outY[mrow * strideOut + 16 + i] = dylo[v];
                }
            }
        }
    }
    acc += lsum * w;
}

__global__ void __launch_bounds__(NTHR, 1)
wavelet_loss_kernel(const float* __restrict__ x, const float* __restrict__ y,
                    float* __restrict__ out, float w1, float w2, float w3)
{
    extern __shared__ float lds[];
    __shared__ float sLo[16], sHi[16];

    const int tid    = threadIdx.x;
    const int lane   = tid & 31;
    const int wave   = tid >> 5;
    const int nwaves = NTHR >> 5;
    const int mrow   = lane & 15;
    const int half   = lane >> 4;

    // Reversed filters: frev_lo[j] = LO[15-j]; frev_hi[j] = (-1)^j * LO[j]
    if (tid < 16) {
        sLo[tid] = cDB8LO[15 - tid];
        sHi[tid] = (tid & 1) ? -cDB8LO[tid] : cDB8LO[tid];
    }
    __syncthreads();

    // Per-wave Toeplitz A operands (same for every level): A[m][k] = frev[k - 2m]
    // A-matrix VGPR layout: lanes 0-15 M=0..15 K={0,1}; lanes 16-31 K={2,3}
    v2f Alo[12], Ahi[12];
#pragma unroll
    for (int kb = 0; kb < 12; ++kb) {
        const int k0 = kb * 4 + 2 * half;
        const int j0 = k0 - 2 * mrow;
        const int j1 = j0 + 1;
        Alo[kb].x = (j0 >= 0 && j0 < 16) ? sLo[j0] : 0.0f;
        Alo[kb].y = (j1 >= 0 && j1 < 16) ? sLo[j1] : 0.0f;
        Ahi[kb].x = (j0 >= 0 && j0 < 16) ? sHi[j0] : 0.0f;
        Ahi[kb].y = (j1 >= 0 && j1 < 16) ? sHi[j1] : 0.0f;
    }

    float* exX1 = lds;                       // 16*ST1 floats
    float* exY1 = lds + 16 * ST1;
    float* exX2 = lds + 32 * ST1;            // 16*ST2 floats
    float* exY2 = exX2 + 16 * ST2;
    float* exX3 = exX1;                      // level-3 aliases dead level-1 space
    float* exY3 = exY1;

    // ---- stage 16 rows of both tensors into LDS with CDNA5 async global->LDS
    // copies (no data VGPRs, tracked by ASYNCcnt; up to 63 outstanding/wave).
    const size_t rowBase = (size_t)blockIdx.x * 16;
    const float* gx = x + rowBase * WTH;
    const float* gy = y + rowBase * WTH;
    for (int idx = tid; idx < 16 * (WTH / 2); idx += NTHR) {   // 8 B chunks, uniform trip count
        const int row = idx >> 9;            // WTH/2 == 512 b64 chunks per row
        const int c2  = idx & 511;
        const unsigned long long gax =
            (unsigned long long)(uintptr_t)(gx + row * WTH + c2 * 2);
        const unsigned long long gay =
            (unsigned long long)(uintptr_t)(gy + row * WTH + c2 * 2);
        // low 32 bits of a shared pointer == LDS byte offset (VDST operand)
        const unsigned ldx = (unsigned)(uintptr_t)(exX1 + row * ST1 + 16 + c2 * 2);
        const unsigned ldy = (unsigned)(uintptr_t)(exY1 + row * ST1 + 16 + c2 * 2);
        asm volatile("global_load_async_to_lds_b64 %0, %1, off"
                     :: "v"(ldx), "v"(gax) : "memory");
        asm volatile("global_load_async_to_lds_b64 %0, %1, off"
                     :: "v"(ldy), "v"(gay) : "memory");
    }
    asm volatile("s_wait_asynccnt 0" ::: "memory");
    __syncthreads();
    fill_pads(exX1, WTH, ST1, tid);
    fill_pads(exY1, WTH, ST1, tid);
    __syncthreads();

    float acc = 0.0f;
    process_level<true >(exX1, exY1, exX2, exY2, M1, ST1, ST2, w1, acc, Alo, Ahi, mrow, half, wave, nwaves);
    __syncthreads();
    fill_pads(exX2, M1, ST2, tid);
    fill_pads(exY2, M1, ST2, tid);
    __syncthreads();
    process_level<true >(exX2, exY2, exX3, exY3, M2, ST2, ST3, w2, acc, Alo, Ahi, mrow, half, wave, nwaves);
    __syncthreads();
    fill_pads(exX3, M2, ST3, tid);
    fill_pads(exY3, M2, ST3, tid);
    __syncthreads();
    process_level<false>(exX3, exY3, nullptr, nullptr, M3, ST3, 0, w3, acc, Alo, Ahi, mrow, half, wave, nwaves);
    __syncthreads();

    // ---- workgroup reduction, one global atomic per workgroup
    float* red = lds;
    red[tid] = acc;
    __syncthreads();
    for (int s = NTHR / 2; s > 0; s >>= 1) {
        if (tid < s) red[tid] += red[tid + s];
        __syncthreads();
    }
    if (tid == 0) atomicAdd(out, red[0]);
}

__global__ void zero_out_kernel(float* p) { *p = 0.0f; }

extern "C" void kernel_launch(void* const* d_in, const int* in_sizes, int n_in,
                              void* d_out, int out_size, void* d_ws, size_t ws_size,
                              hipStream_t stream)
{
    (void)n_in; (void)out_size; (void)d_ws; (void)ws_size;
    const float* x = (const float*)d_in[0];
    const float* y = (const float*)d_in[1];
    float* out = (float*)d_out;

    const long long total = (long long)in_sizes[0];   // B*C*H*W
    const long long R     = total / WTH;              // rows
    const int grid        = (int)(R / 16);

    const float w1 = 1.0f / (3.0f * (float)R * (float)M1);
    const float w2 = 1.0f / (3.0f * (float)R * (float)M2);
    const float w3 = 1.0f / (3.0f * (float)R * (float)M3);

    const size_t shmem = (size_t)(32 * ST1 + 32 * ST2) * sizeof(float); // ~205 KB / WGP
    (void)hipFuncSetAttribute((const void*)wavelet_loss_kernel,
                              hipFuncAttributeMaxDynamicSharedMemorySize, (int)shmem);

    zero_out_kernel<<<1, 1, 0, stream>>>(out);
    wavelet_loss_kernel<<<grid, NTHR, shmem, stream>>>(x, y, out, w1, w2, w3);
}